// CLIPVisionTower_VisionZip_22204980920418
// MI455X (gfx1250) — compile-verified
//
#include <hip/hip_runtime.h>
#include <float.h>
#include <math.h>

typedef __attribute__((ext_vector_type(2))) float v2f;
typedef __attribute__((ext_vector_type(8))) float v8f;

#define BATCH  16
#define HEADS  16
#define SEQ    577
#define DIM    1024
#define DM     64
#define DOM    54            // dominant_num
#define NSEL   (DOM + 1)     // 55 (cls + dominant)
#define NKEEP  (SEQ - NSEL)  // 522
#define NCTX   10            // contextual_num
#define STEP   52            // NKEEP / NCTX
#define NMERGE (NKEEP - NCTX) // 512

// Map merge-token index m in [0,512) to its position in the kept list.
// Targets sit at kept positions {0,52,...,468}; between consecutive targets
// there are 51 merge tokens, and 53 after the last target.
__device__ __forceinline__ int merge_pos(int m) {
    if (m < 9 * 51) { int t = m / 51; int r = m - t * 51; return t * STEP + 1 + r; }
    return m + NCTX; // 459 -> 469 ... 511 -> 521
}

// ---------------------------------------------------------------------------
// Kernel 1: per-batch CLS attention head-sum, iterative top-54 (stable,
// lowest-index tie-break like lax.top_k), build sorted dominant + kept lists.
// ---------------------------------------------------------------------------
__global__ void __launch_bounds__(256) k_topk_select(
    const float* __restrict__ attn,   // [B][H][S][S]
    int* __restrict__ all_idx,        // [B][55] selection order
    int* __restrict__ dom_idx,        // [B][55] ascending
    int* __restrict__ kept_idx,       // [B][522] ascending
    float* __restrict__ out_idx)      // d_out tail: all_indices as float
{
    const int b = blockIdx.x;
    const int tid = threadIdx.x;

    __shared__ float sums[SEQ - 1];
    __shared__ float bval[256];
    __shared__ int   bidx[256];
    __shared__ int   topk[DOM];
    __shared__ int   flags[SEQ];

    // cls_attention_sum[j] = sum_h attn[b,h,0,j+1]
    for (int jj = tid; jj < SEQ - 1; jj += 256) {
        float s = 0.f;
        size_t base = (size_t)b * HEADS * SEQ * SEQ + (size_t)(jj + 1);
        for (int h = 0; h < HEADS; ++h)
            s += attn[base + (size_t)h * SEQ * SEQ];
        sums[jj] = s;
    }
    __syncthreads();

    // 54 rounds of block-parallel argmax with lowest-index tie-break
    for (int k = 0; k < DOM; ++k) {
        float v = -FLT_MAX; int id = SEQ;
        for (int jj = tid; jj < SEQ - 1; jj += 256) {
            float s = sums[jj];
            if (s > v) { v = s; id = jj; }   // ascending jj: ties keep first
        }
        bval[tid] = v; bidx[tid] = id;
        __syncthreads();
        if (tid == 0) {
            float bv = bval[0]; int bi = bidx[0];
            for (int t = 1; t < 256; ++t) {
                if (bval[t] > bv || (bval[t] == bv && bidx[t] < bi)) {
                    bv = bval[t]; bi = bidx[t];
                }
            }
            topk[k] = bi;
            sums[bi] = -FLT_MAX;
        }
        __syncthreads();
    }

    for (int i = tid; i < SEQ; i += 256) flags[i] = 0;
    __syncthreads();

    if (tid == 0) {
        flags[0] = 1;
        all_idx[b * NSEL + 0] = 0;
        out_idx[b * NSEL + 0] = 0.0f;
        for (int k = 0; k < DOM; ++k) {
            int tok = topk[k] + 1;
            flags[tok] = 1;
            all_idx[b * NSEL + k + 1] = tok;
            out_idx[b * NSEL + k + 1] = (float)tok;
        }
        int nd = 0, nk = 0;
        for (int i = 0; i < SEQ; ++i) {
            if (flags[i]) dom_idx[b * NSEL + nd++] = i;
            else          kept_idx[b * NKEEP + nk++] = i;
        }
    }
}

// ---------------------------------------------------------------------------
// Kernel 2: reciprocal L2 norm of metric rows for kept tokens.
// ---------------------------------------------------------------------------
__global__ void k_rnorm(const float* __restrict__ metric,
                        const int* __restrict__ kept_idx,
                        float* __restrict__ rnorm)
{
    int g = blockIdx.x * blockDim.x + threadIdx.x;
    if (g >= BATCH * NKEEP) return;
    int b = g / NKEEP, p = g - b * NKEEP;
    int orig = kept_idx[b * NKEEP + p];
    const float* row = metric + ((size_t)b * SEQ + orig) * DM;
    float s = 0.f;
    for (int k = 0; k < DM; ++k) { float x = row[k]; s += x * x; }
    rnorm[g] = 1.0f / sqrtf(s);
}

// ---------------------------------------------------------------------------
// Kernel 3: similarity GEMM via V_WMMA_F32_16X16X4_F32 + per-row argmax.
// One block per batch, 8 waves x 4 tiles = 32 tiles of 16 rows = 512 merge
// tokens. N padded to 16 (rows 10..15 zero); argmax scans only n<10.
// A layout (16x4 f32): lanes 0-15 rows, VGPR0/1 = K{0,1}; lanes 16-31 K{2,3}.
// ---------------------------------------------------------------------------
__global__ void __launch_bounds__(256) k_similarity_assign(
    const float* __restrict__ metric,
    const int* __restrict__ kept_idx,
    const float* __restrict__ rnorm,
    int* __restrict__ assign)         // [B][512]
{
    const int b = blockIdx.x;
    const int tid = threadIdx.x;
    const int wave = tid >> 5;
    const int lane = tid & 31;
    const int half = lane >> 4;
    const int l15  = lane & 15;

    __shared__ float Tsh[16 * DM];   // normalized targets, [n][k], rows 10..15 = 0
    __shared__ float Dsh[8][256];    // per-wave 16x16 result tile

    for (int idx = tid; idx < 16 * DM; idx += 256) {
        int n = idx >> 6, k = idx & 63;
        float v = 0.f;
        if (n < NCTX) {
            int p = n * STEP;
            int orig = kept_idx[b * NKEEP + p];
            v = metric[((size_t)b * SEQ + orig) * DM + k] * rnorm[b * NKEEP + p];
        }
        Tsh[idx] = v;
    }
    __syncthreads();

    for (int it = 0; it < 4; ++it) {
        const int tile = wave + 8 * it;          // 0..31
        const int mrow = tile * 16 + l15;        // merge index for this lane's row
        const int p    = merge_pos(mrow);
        const int orig = kept_idx[b * NKEEP + p];
        const float rn = rnorm[b * NKEEP + p];
        const float* arow = metric + ((size_t)b * SEQ + orig) * DM;

        v8f acc = {0.f, 0.f, 0.f, 0.f, 0.f, 0.f, 0.f, 0.f};
        #pragma unroll
        for (int ks = 0; ks < 16; ++ks) {
            const int k0 = ks * 4 + 2 * half;
            v2f a, bb;
            a.x  = arow[k0]     * rn;
            a.y  = arow[k0 + 1] * rn;
            bb.x = Tsh[l15 * DM + k0];
            bb.y = Tsh[l15 * DM + k0 + 1];
            acc = __builtin_amdgcn_wmma_f32_16x16x4_f32(
                false, a, false, bb, (short)0, acc, false, false);
        }

        // D layout: VGPR v -> row (v + 8*half), col l15
        #pragma unroll
        for (int v = 0; v < 8; ++v)
            Dsh[wave][(v + 8 * half) * 16 + l15] = acc[v];
        __syncthreads();

        if (half == 0) {   // 16 lanes: one row each; first-max tie-break = argmax
            float best = Dsh[wave][l15 * 16 + 0]; int bi = 0;
            for (int n = 1; n < NCTX; ++n) {
                float s = Dsh[wave][l15 * 16 + n];
                if (s > best) { best = s; bi = n; }
            }
            assign[b * NMERGE + tile * 16 + l15] = bi;
        }
        __syncthreads();
    }
}

// ---------------------------------------------------------------------------
// Kernel 4: counts, 10x1024 LDS aggregation (thread-owns-dim: conflict free),
// dominant gather + contextual token write.
// ---------------------------------------------------------------------------
__global__ void __launch_bounds__(256) k_aggregate(
    const float* __restrict__ hidden,
    const int* __restrict__ kept_idx,
    const int* __restrict__ dom_idx,
    const int* __restrict__ assign,
    float* __restrict__ out)          // [B][65][1024]
{
    const int b = blockIdx.x;
    const int tid = threadIdx.x;

    __shared__ float agg[NCTX * DIM];   // 40 KB (WGP has 320 KB LDS)
    __shared__ int   assign_s[NMERGE];
    __shared__ int   morig[NMERGE];
    __shared__ int   counts[NCTX];
    __shared__ int   tgtorig[NCTX];
    __shared__ int   dom_s[NSEL];

    if (tid < NCTX) counts[tid] = 0;
    for (int i = tid; i < NCTX * DIM; i += 256) agg[i] = 0.f;
    __syncthreads();

    for (int i = tid; i < NMERGE; i += 256) {
        int c = assign[b * NMERGE + i];
        assign_s[i] = c;
        atomicAdd(&counts[c], 1);
        morig[i] = kept_idx[b * NKEEP + merge_pos(i)];
    }
    if (tid < NCTX) tgtorig[tid] = kept_idx[b * NKEEP + tid * STEP];
    for (int i = tid; i < NSEL; i += 256) dom_s[i] = dom_idx[b * NSEL + i];
    __syncthreads();

    const float* hb = hidden + (size_t)b * SEQ * DIM;
    float* ob = out + (size_t)b * (NSEL + NCTX) * DIM;

    // dominant tokens (sorted index gather)
    for (int i = 0; i < NSEL; ++i) {
        const float* src = hb + (size_t)dom_s[i] * DIM;
        #pragma unroll
        for (int it = 0; it < 4; ++it) {
            int d = tid + 256 * it;
            ob[i * DIM + d] = src[d];
        }
    }

    // scatter-sum: each thread owns dims {tid, tid+256, tid+512, tid+768}
    #pragma unroll
    for (int it = 0; it < 4; ++it) {
        int d = tid + 256 * it;
        for (int m = 0; m < NMERGE; ++m)
            agg[assign_s[m] * DIM + d] += hb[(size_t)morig[m] * DIM + d];
    }
    __syncthreads();

    // contextual tokens
    for (int c = 0; c < NCTX; ++c) {
        float cnt = fmaxf((float)counts[c], 1.0f);
        const float* src = hb + (size_t)tgtorig[c] * DIM;
        #pragma unroll
        for (int it = 0; it < 4; ++it) {
            int d = tid + 256 * it;
            ob[(NSEL + c) * DIM + d] = src[d] + agg[c * DIM + d] / cnt;
        }
    }
}

// ---------------------------------------------------------------------------
extern "C" void kernel_launch(void* const* d_in, const int* in_sizes, int n_in,
                              void* d_out, int out_size, void* d_ws, size_t ws_size,
                              hipStream_t stream)
{
    (void)in_sizes; (void)n_in; (void)out_size; (void)ws_size;

    const float* attn   = (const float*)d_in[0];  // (16,16,577,577)
    const float* hidden = (const float*)d_in[1];  // (16,577,1024)
    const float* metric = (const float*)d_in[2];  // (16,577,64)
    // d_in[3]=dominant_num(54), d_in[4]=contextual_num(10): baked as constants.

    float* out = (float*)d_out;
    float* out_idx = out + (size_t)BATCH * (NSEL + NCTX) * DIM; // indices tail (as float)

    // workspace layout (bytes, 64B-aligned sections)
    char* ws = (char*)d_ws;
    int*   all_idx  = (int*)(ws + 0);                          // 16*55*4   = 3520
    int*   dom_idx  = (int*)(ws + 3584);                       // 3520
    int*   kept_idx = (int*)(ws + 7168);                       // 16*522*4  = 33408
    float* rnorm    = (float*)(ws + 7168 + 33408);             // 33408
    int*   assign   = (int*)(ws + 7168 + 33408 + 33408);       // 16*512*4  = 32768
    // total ~104 KB

    k_topk_select<<<BATCH, 256, 0, stream>>>(attn, all_idx, dom_idx, kept_idx, out_idx);
    k_rnorm<<<(BATCH * NKEEP + 255) / 256, 256, 0, stream>>>(metric, kept_idx, rnorm);
    k_similarity_assign<<<BATCH, 256, 0, stream>>>(metric, kept_idx, rnorm, assign);
    k_aggregate<<<BATCH, 256, 0, stream>>>(hidden, kept_idx, dom_idx, assign, out);
}